// MF_target_76673756168567
// MI455X (gfx1250) — compile-verified
//
#include <hip/hip_runtime.h>
#include <hip/hip_bf16.h>

#define EMBED_DIM 64
#define BATCH     2048
#define NTOT      4096        // 2*BATCH
#define NUM_ITEMS 10000
#define NTILE     (NTOT / 16) // 256 tiles per side

typedef __attribute__((ext_vector_type(2))) float v2f;
typedef __attribute__((ext_vector_type(8))) float v8f;

__device__ __forceinline__ float wave_reduce_add(float v) {
#pragma unroll
    for (int off = 16; off >= 1; off >>= 1)
        v += __shfl_xor(v, off, 32);
    return v;
}

// ---------------------------------------------------------------------------
// Kernel 1: gather e = item_table[total], row squared norms, ids; zero d_out.
// One wave (32 lanes) per row; lane l handles dims l and l+32.
// ---------------------------------------------------------------------------
__global__ void prep_kernel(const int* __restrict__ pos_id,
                            const int* __restrict__ neg_id,
                            const float* __restrict__ item_table,
                            float* __restrict__ eW,
                            float* __restrict__ sqW,
                            int* __restrict__ tidW,
                            float* __restrict__ out) {
    if (blockIdx.x == 0 && threadIdx.x < 2) out[threadIdx.x] = 0.0f;

    const int row  = (int)((blockIdx.x * blockDim.x + threadIdx.x) >> 5);
    const int lane = (int)(threadIdx.x & 31);
    if (row >= NTOT) return;

    const int t = (row < BATCH) ? pos_id[row] : neg_id[row - BATCH];
    const float* src = item_table + (size_t)t * EMBED_DIM;
    const float a = src[lane];
    const float b = src[lane + 32];
    eW[(size_t)row * EMBED_DIM + lane]      = a;
    eW[(size_t)row * EMBED_DIM + lane + 32] = b;

    const float s = wave_reduce_add(a * a + b * b);
    if (lane == 0) { sqW[row] = s; tidW[row] = t; }
}

// ---------------------------------------------------------------------------
// Kernel 2: BPR softplus loss. One wave per batch row.
// ---------------------------------------------------------------------------
__global__ void bpr_loss_kernel(const int* __restrict__ user_id,
                                const int* __restrict__ pos_id,
                                const int* __restrict__ neg_id,
                                const float* __restrict__ user_table,
                                const float* __restrict__ item_table,
                                float* __restrict__ out) {
    const int row  = (int)((blockIdx.x * blockDim.x + threadIdx.x) >> 5);
    const int lane = (int)(threadIdx.x & 31);
    if (row >= BATCH) return;

    const float* ue = user_table + (size_t)user_id[row] * EMBED_DIM;
    const float* pe = item_table + (size_t)pos_id[row]  * EMBED_DIM;
    const float* ne = item_table + (size_t)neg_id[row]  * EMBED_DIM;

    const float u0 = ue[lane], u1 = ue[lane + 32];
    float ps = u0 * pe[lane] + u1 * pe[lane + 32];
    float ns = u0 * ne[lane] + u1 * ne[lane + 32];
    ps = wave_reduce_add(ps);
    ns = wave_reduce_add(ns);

    if (lane == 0) {
        const float x  = ns - ps;
        // numerically stable softplus: max(x,0) + log1p(exp(-|x|))
        const float sp = fmaxf(x, 0.0f) + log1pf(expf(-fabsf(x)));
        atomicAdd(out, sp);
    }
}

// ---------------------------------------------------------------------------
// Kernel 3: fused Gram + distance + Dg gather + masked reduction.
// One wave computes one 16x16 tile of e@e^T via 16x V_WMMA_F32_16X16X4_F32,
// then applies mask / sqrt / Dg gather and reduces to a single atomicAdd.
// Only upper-triangular tiles (J >= I) are processed (uniform branch, so
// EXEC is all-1s at every WMMA as required).
//
// fp32 WMMA operand layout (ISA 7.12.2):
//   A (16x4): lanes 0-15 -> M=lane, VGPR0/1 = K={k0,k0+1};
//             lanes 16-31 -> M=lane-16, K={k0+2,k0+3}   => float2 per lane.
//   B (4x16): mirrored with N in lanes (same load pattern, tile J).
//   C/D (16x16, 8 VGPRs): VGPR r -> M = r + 8*(lane>=16), N = lane&15.
// ---------------------------------------------------------------------------
__global__ void reg_kernel(const float* __restrict__ eW,
                           const float* __restrict__ sqW,
                           const int* __restrict__ tidW,
                           const float* __restrict__ Dmat,
                           float* __restrict__ out) {
    const int lane  = (int)(threadIdx.x & 31);
    const int waveB = (int)(threadIdx.x >> 5);
    const int J = (int)blockIdx.x * 8 + waveB;  // column tile
    const int I = (int)blockIdx.y;              // row tile
    if (J < I) return;                          // wave-uniform skip

    const int half = lane >> 4;   // 0: lanes 0-15, 1: lanes 16-31
    const int l15  = lane & 15;

    const float* arow = eW + (size_t)(I * 16 + l15) * EMBED_DIM + 2 * half;
    const float* brow = eW + (size_t)(J * 16 + l15) * EMBED_DIM + 2 * half;

    v8f c = {};
#pragma unroll
    for (int k0 = 0; k0 < EMBED_DIM; k0 += 4) {
        const v2f a = *(const v2f*)(arow + k0);
        const v2f b = *(const v2f*)(brow + k0);
        // (neg_a, A, neg_b, B, c_mod, C, reuse_a, reuse_b)
        c = __builtin_amdgcn_wmma_f32_16x16x4_f32(
                false, a, false, b, (short)0, c, false, false);
    }

    // Per-lane column-side data (fixed across the 8 accumulator VGPRs).
    const int   j   = J * 16 + l15;
    const int   jt  = tidW[j];
    const float sqj = sqW[j];

    float acc = 0.0f;
#pragma unroll
    for (int r = 0; r < 8; ++r) {
        const int m = r + 8 * half;
        const int i = I * 16 + m;
        const float d2 = sqW[i] + sqj - 2.0f * c[r];
        if (j > i && d2 > 0.0f) {
            const float dg = Dmat[(size_t)tidW[i] * NUM_ITEMS + jt];
            acc += dg * sqrtf(d2);
        }
    }
    acc = wave_reduce_add(acc);
    if (lane == 0) atomicAdd(out + 1, acc);
}

// ---------------------------------------------------------------------------
extern "C" void kernel_launch(void* const* d_in, const int* in_sizes, int n_in,
                              void* d_out, int out_size, void* d_ws, size_t ws_size,
                              hipStream_t stream) {
    const int*   user_id    = (const int*)d_in[0];
    const int*   pos_id     = (const int*)d_in[1];
    const int*   neg_id     = (const int*)d_in[2];
    const float* user_table = (const float*)d_in[3];
    const float* item_table = (const float*)d_in[4];
    const float* Dmat       = (const float*)d_in[5];
    float*       out        = (float*)d_out;

    // Workspace layout: e (4096x64 f32) | sq (4096 f32) | tid (4096 i32) ~ 1.08 MB
    float* eW   = (float*)d_ws;
    float* sqW  = eW + (size_t)NTOT * EMBED_DIM;
    int*   tidW = (int*)(sqW + NTOT);

    // 8 waves per 256-thread block.
    prep_kernel<<<NTOT / 8, 256, 0, stream>>>(pos_id, neg_id, item_table,
                                              eW, sqW, tidW, out);
    bpr_loss_kernel<<<BATCH / 8, 256, 0, stream>>>(user_id, pos_id, neg_id,
                                                   user_table, item_table, out);
    dim3 grid(NTILE / 8, NTILE); // wave w of block (x,y) -> tile (I=y, J=8x+w)
    reg_kernel<<<grid, 256, 0, stream>>>(eW, sqW, tidW, Dmat, out);
}